// KANLinear_8701603741760
// MI455X (gfx1250) — compile-verified
//
#include <hip/hip_runtime.h>
#include <hip/hip_bf16.h>
#include <math.h>

typedef __attribute__((ext_vector_type(2))) float v2f;
typedef __attribute__((ext_vector_type(8))) float v8f;

#define IN_F   256
#define OUT_F  256
#define BATCH  1024
#define NBASIS 7          // grid_size + order - 1 = 5 + 3 - 1
#define NB_PAD 8          // padded for WMMA K and aligned float2 loads

__device__ __forceinline__ float sdiv(float n, float d) {
    return (d != 0.0f) ? (n / d) : 0.0f;
}

// ---------------------------------------------------------------------------
// Kernel 1: B-spline basis (degree 2 after the d=1..k-1 recursion), one thread
// per (b, i). Mirrors reference: eg = [L,L,L,g0..g4,R,R,R], L=2g0-g3, R=2g4-g1.
// Writes basis[b][i][0..6], zero at [7] (WMMA K-padding).
// ---------------------------------------------------------------------------
__global__ void kan_basis_kernel(const float* __restrict__ x,
                                 const float* __restrict__ grid,
                                 float* __restrict__ basis) {
    const int idx = blockIdx.x * blockDim.x + threadIdx.x;   // 0 .. 1024*256-1
    const int i = idx & (IN_F - 1);
    const int b = idx >> 8;
    const float xv = x[b * IN_F + i];

    float g[5];
#pragma unroll
    for (int j = 0; j < 5; ++j) g[j] = grid[j * IN_F + i];
    const float L = 2.0f * g[0] - g[3];
    const float R = 2.0f * g[4] - g[1];

    float eg[11] = {L, L, L, g[0], g[1], g[2], g[3], g[4], R, R, R};

    // degree 0: B0[j] = [eg[j] <= x < eg[j+1]], j=0..9
    float B0[10];
#pragma unroll
    for (int j = 0; j < 10; ++j)
        B0[j] = (xv >= eg[j] && xv < eg[j + 1]) ? 1.0f : 0.0f;

    // degree 1 (d=1): j=0..8
    float B1[9];
#pragma unroll
    for (int j = 0; j < 9; ++j) {
        float lt = sdiv(xv - eg[j],     eg[j + 1] - eg[j]);
        float rt = sdiv(eg[j + 2] - xv, eg[j + 2] - eg[j + 1]);
        B1[j] = lt * B0[j] + rt * B0[j + 1];
    }

    // degree 2 (d=2): j=0..7, keep first 7
    float* bp = basis + (size_t)(b * IN_F + i) * NB_PAD;
#pragma unroll
    for (int j = 0; j < 7; ++j) {
        float lt = sdiv(xv - eg[j],     eg[j + 2] - eg[j]);
        float rt = sdiv(eg[j + 3] - xv, eg[j + 3] - eg[j + 1]);
        bp[j] = lt * B1[j] + rt * B1[j + 1];
    }
    bp[7] = 0.0f;
}

// ---------------------------------------------------------------------------
// Kernel 2: base = x @ W via V_WMMA_F32_16X16X4_F32, one 16x16 tile per wave,
// K = 256 -> 64 WMMA steps. 1024 wave-tiles = 128 blocks of 8 waves.
// A layout (16x4 f32): lane half selects K pair {0,1} / {2,3}.
// ---------------------------------------------------------------------------
__global__ void kan_base_kernel(const float* __restrict__ x,
                                const float* __restrict__ w,
                                float* __restrict__ base_out) {
    const int lane = threadIdx.x & 31;
    const int wid  = threadIdx.x >> 5;
    const int wg   = blockIdx.x * 8 + wid;        // 0..1023
    const int otile = wg & 15;
    const int btile = wg >> 4;                    // 0..63
    const int hh = lane >> 4;                     // lane half
    const int m  = lane & 15;
    const int b0 = btile * 16, o0 = otile * 16;

    v8f c = {};
    for (int kc = 0; kc < 64; ++kc) {
        const int k0 = kc * 4 + 2 * hh;           // {0,2} + 4*kc per half
        v2f a, bb;
        a.x  = x[(b0 + m) * IN_F + k0];
        a.y  = x[(b0 + m) * IN_F + k0 + 1];
        bb.x = w[(size_t)k0 * OUT_F + (o0 + m)];
        bb.y = w[(size_t)(k0 + 1) * OUT_F + (o0 + m)];
        c = __builtin_amdgcn_wmma_f32_16x16x4_f32(false, a, false, bb,
                                                  (short)0, c, false, false);
    }
#pragma unroll
    for (int v = 0; v < 8; ++v)                   // C/D: M = v + 8*half, N = m
        base_out[(b0 + v + 8 * hh) * OUT_F + (o0 + m)] = c[v];
}

// ---------------------------------------------------------------------------
// Kernel 3: fused spline WMMA + base broadcast + SiLU.
// Wave owns a (b-tile, o-tile), loops 32 values of i; base tile lives in regs.
// Per i: K=8 (7 real) -> 2 WMMA f32 16x16x4 steps, epilogue silu, store.
// 8192 waves = 1024 blocks of 256 threads.
// ---------------------------------------------------------------------------
__global__ void kan_main_kernel(const float* __restrict__ basis,   // [1024][256][8]
                                const float* __restrict__ sw,      // [256][256][7]
                                const float* __restrict__ base_out,// [1024][256]
                                float* __restrict__ out) {         // [1024][256][256]
    const int lane = threadIdx.x & 31;
    const int wid  = threadIdx.x >> 5;
    const int wg   = blockIdx.x * 8 + wid;        // 0..8191
    const int otile  = wg & 15;
    const int btile  = (wg >> 4) & 63;
    const int ichunk = wg >> 10;                  // 0..7
    const int hh = lane >> 4;
    const int m  = lane & 15;
    const int b0 = btile * 16, o0 = otile * 16;

    // base tile (16x16) resident in registers: M = v + 8*hh, N = m
    float basev[8];
#pragma unroll
    for (int v = 0; v < 8; ++v)
        basev[v] = base_out[(b0 + v + 8 * hh) * OUT_F + (o0 + m)];

    const int arow = b0 + m;                      // A-matrix row = batch index

    for (int ii = 0; ii < 32; ++ii) {
        const int i = ichunk * 32 + ii;
        v8f c = {};

        // K chunk 0: k = {0,1} / {2,3} by lane half
        {
            const float* ap = basis + ((size_t)arow * IN_F + i) * NB_PAD + 2 * hh;
            v2f a;  a.x = ap[0]; a.y = ap[1];
            const float* bp = sw + ((size_t)i * OUT_F + (o0 + m)) * NBASIS + 2 * hh;
            v2f bb; bb.x = bp[0]; bb.y = bp[1];
            c = __builtin_amdgcn_wmma_f32_16x16x4_f32(false, a, false, bb,
                                                      (short)0, c, false, false);
        }
        // K chunk 1: k = {4,5} / {6,7}; k=7 is zero (basis padded, sw guarded)
        {
            const float* ap = basis + ((size_t)arow * IN_F + i) * NB_PAD + 4 + 2 * hh;
            v2f a;  a.x = ap[0]; a.y = ap[1];     // ap[1] is the 0.0 pad when hh==1
            const int k0 = 4 + 2 * hh;
            const float* bp = sw + ((size_t)i * OUT_F + (o0 + m)) * NBASIS + k0;
            v2f bb; bb.x = bp[0];
            bb.y = (k0 + 1 < NBASIS) ? bp[1] : 0.0f;
            c = __builtin_amdgcn_wmma_f32_16x16x4_f32(false, a, false, bb,
                                                      (short)0, c, false, false);
        }

        // epilogue: add base, SiLU, store (row of 16 consecutive o per VGPR)
#pragma unroll
        for (int v = 0; v < 8; ++v) {
            float s = c[v] + basev[v];
            float r = s / (1.0f + __expf(-s));
            const size_t brow = (size_t)(b0 + v + 8 * hh);
            out[(brow * IN_F + i) * OUT_F + (o0 + m)] = r;
        }
    }
}

// ---------------------------------------------------------------------------
extern "C" void kernel_launch(void* const* d_in, const int* in_sizes, int n_in,
                              void* d_out, int out_size, void* d_ws, size_t ws_size,
                              hipStream_t stream) {
    const float* x    = (const float*)d_in[0];   // (1024, 256)
    const float* bw   = (const float*)d_in[1];   // (256, 256)
    const float* sw   = (const float*)d_in[2];   // (256, 256, 7)
    const float* grid = (const float*)d_in[3];   // (5, 256)
    float* out = (float*)d_out;                  // (1024, 256, 256)

    float* basis    = (float*)d_ws;                          // 1024*256*8 f32 = 8 MB
    float* base_out = basis + (size_t)BATCH * IN_F * NB_PAD; // 1024*256 f32  = 1 MB

    kan_basis_kernel<<<(BATCH * IN_F) / 256, 256, 0, stream>>>(x, grid, basis);
    kan_base_kernel<<<128, 256, 0, stream>>>(x, bw, base_out);
    kan_main_kernel<<<1024, 256, 0, stream>>>(basis, sw, base_out, out);
}